// ODEFunc_24653112279323
// MI455X (gfx1250) — compile-verified
//
#include <hip/hip_runtime.h>
#include <math.h>

// ---------------------------------------------------------------------------
// Problem constants (from the reference)
// ---------------------------------------------------------------------------
constexpr int Nn = 256;           // nodes
constexpr int Vv = 64;            // node feature dim
constexpr int Ee = 64;            // edge feature dim
constexpr int Ss = 16;            // leading batch (time) dim
constexpr int Hh = 16;            // hidden dim
constexpr int NE = Nn * (Nn - 1); // 65280 edges per batch slice
constexpr int TILES_EDGE = (Ss * NE) / 16;  // 65280 wave-tiles
constexpr int TILES_NODE = (Ss * Nn) / 16;  // 256 wave-tiles

typedef __attribute__((ext_vector_type(2))) float v2f;
typedef __attribute__((ext_vector_type(8))) float v8f;

// D(16x16 f32) = A(16x4 f32) * B(4x16 f32) + C
// A layout: lane m=L%16 holds K=kb+{0,1} (lanes 0-15) or kb+{2,3} (lanes 16-31)
// B layout: lane n=L%16 holds the mirrored K striping for column n
// C/D layout: lane n=L%16 holds rows r (lanes 0-15) / r+8 (lanes 16-31), r=VGPR 0..7
static __device__ __forceinline__ v8f wmma_f32(v2f a, v2f b, v8f c) {
  return __builtin_amdgcn_wmma_f32_16x16x4_f32(
      /*neg_a=*/false, a, /*neg_b=*/false, b,
      /*c_mod=*/(short)0, c, /*reuse_a=*/false, /*reuse_b=*/false);
}

// ---------------------------------------------------------------------------
// Kernel 1: ev[s,n,:] = sum_j e0[s,n,j,:]   (streams the 267 MB e0 tensor once)
// One block per (s,n); 255*64 = 16320 contiguous floats reduced to 64 channels.
// Result written into the dv slot of d_out (consumed in-place by kernel 2).
// ---------------------------------------------------------------------------
__global__ __launch_bounds__(256) void k_sum_e0(const float* __restrict__ e0,
                                                float* __restrict__ ev) {
  const int b = blockIdx.x;   // s*Nn + n
  const int t = threadIdx.x;
  const float* src = e0 + (size_t)b * (Nn - 1) * Ee;

  float4 acc = {0.f, 0.f, 0.f, 0.f};
  for (int idx = 4 * t; idx < (Nn - 1) * Ee; idx += 1024) {  // stride ≡ 0 mod 64
    float4 x = *(const float4*)(src + idx);
    acc.x += x.x; acc.y += x.y; acc.z += x.z; acc.w += x.w;
  }

  __shared__ float4 red[256];
  red[t] = acc;
  __syncthreads();
  // thread t always touches channels 4*(t%16)..+3 -> reduce groups mod 16
  for (int sft = 128; sft >= 16; sft >>= 1) {
    if (t < sft) {
      float4 o = red[t + sft];
      red[t].x += o.x; red[t].y += o.y; red[t].z += o.z; red[t].w += o.w;
    }
    __syncthreads();
  }
  if (t < 16) *(float4*)(ev + (size_t)b * Ee + 4 * t) = red[t];
}

// ---------------------------------------------------------------------------
// Kernel 2: dv = tanh(ev @ wv1 + bv1) @ wv2 + bv2   (in-place on the ev rows)
// One wave per 16-row tile. 16 + 16 WMMAs per tile.
// ---------------------------------------------------------------------------
__global__ __launch_bounds__(256) void k_node_mlp(float* __restrict__ ev,
                                                  const float* __restrict__ wv1,
                                                  const float* __restrict__ bv1,
                                                  const float* __restrict__ wv2,
                                                  const float* __restrict__ bv2) {
  const int lane = threadIdx.x & 31;
  const int w    = threadIdx.x >> 5;
  const int tile = blockIdx.x * 8 + w;        // 0..255, exact
  const int m    = lane & 15;
  const int off  = (lane < 16) ? 0 : 2;
  const int hi8  = (lane < 16) ? 0 : 8;

  const float* row = ev + (size_t)(tile * 16 + m) * Ee;

  // ---- stage 1: 16x64 @ 64x16 ----
  v8f acc = {0.f, 0.f, 0.f, 0.f, 0.f, 0.f, 0.f, 0.f};
  for (int kb = 0; kb < Ee; kb += 4) {
    v2f a; a.x = row[kb + off];            a.y = row[kb + off + 1];
    v2f b; b.x = wv1[(kb + off) * Hh + m]; b.y = wv1[(kb + off + 1) * Hh + m];
    acc = wmma_f32(a, b, acc);
  }

  // tanh + bias, then transpose D-layout -> A-layout through LDS
  __shared__ float hs[8][16][17];
  const float bias1 = bv1[m];
#pragma unroll
  for (int r = 0; r < 8; ++r) hs[w][hi8 + r][m] = tanhf(acc[r] + bias1);
  __syncthreads();

  v2f a2[4];
#pragma unroll
  for (int kb = 0; kb < Hh; kb += 4) {
    a2[kb >> 2].x = hs[w][m][kb + off];
    a2[kb >> 2].y = hs[w][m][kb + off + 1];
  }

  // ---- stage 2: 16x16 @ 16x64 (+ bias), 4 column tiles ----
  for (int nt = 0; nt < 4; ++nt) {
    const int nc = nt * 16 + m;
    const float bias2 = bv2[nc];
    v8f c;
#pragma unroll
    for (int r = 0; r < 8; ++r) c[r] = bias2;
#pragma unroll
    for (int kb = 0; kb < Hh; kb += 4) {
      v2f b; b.x = wv2[(kb + off) * Vv + nc]; b.y = wv2[(kb + off + 1) * Vv + nc];
      c = wmma_f32(a2[kb >> 2], b, c);
    }
    // in-place store: all reads of these rows already completed (dataflow order)
#pragma unroll
    for (int r = 0; r < 8; ++r)
      ev[(size_t)(tile * 16 + hi8 + r) * Vv + nc] = c[r];
  }
}

// ---------------------------------------------------------------------------
// Kernel 3: de = tanh(concat(v0[recv], v0[send]) @ we1 + be1) @ we2 + be2
// One wave per (s, 16-edge tile): 32 + 16 WMMAs, 4 KB of output staged in LDS
// (row stride 68 -> conflict-free writes) and stored as coalesced float4s.
// ---------------------------------------------------------------------------
__global__ __launch_bounds__(256) void k_edge_mlp(const float* __restrict__ v0,
                                                  const float* __restrict__ we1,
                                                  const float* __restrict__ be1,
                                                  const float* __restrict__ we2,
                                                  const float* __restrict__ be2,
                                                  const int* __restrict__ recv,
                                                  const int* __restrict__ send,
                                                  float* __restrict__ de) {
  const int lane = threadIdx.x & 31;
  const int w    = threadIdx.x >> 5;
  const int g    = blockIdx.x * 8 + w;   // 0..65279, exact
  const int s    = g / (NE / 16);
  const int et   = g % (NE / 16);
  const int m    = lane & 15;
  const int off  = (lane < 16) ? 0 : 2;
  const int hi8  = (lane < 16) ? 0 : 8;

  const int j = et * 16 + m;
  const float* rrow = v0 + ((size_t)s * Nn + recv[j]) * Vv;  // hot in L2 (1 MB table)
  const float* srow = v0 + ((size_t)s * Nn + send[j]) * Vv;

  // ---- stage 1: 16x128 @ 128x16 (K split: recv half then send half) ----
  v8f acc = {0.f, 0.f, 0.f, 0.f, 0.f, 0.f, 0.f, 0.f};
#pragma unroll 4
  for (int kb = 0; kb < Vv; kb += 4) {
    v2f a; a.x = rrow[kb + off];            a.y = rrow[kb + off + 1];
    v2f b; b.x = we1[(kb + off) * Hh + m];  b.y = we1[(kb + off + 1) * Hh + m];
    acc = wmma_f32(a, b, acc);
  }
#pragma unroll 4
  for (int kb = 0; kb < Vv; kb += 4) {
    v2f a; a.x = srow[kb + off];                 a.y = srow[kb + off + 1];
    v2f b; b.x = we1[(Vv + kb + off) * Hh + m];  b.y = we1[(Vv + kb + off + 1) * Hh + m];
    acc = wmma_f32(a, b, acc);
  }

  __shared__ float hs[8][16][17];
  const float bias1 = be1[m];
#pragma unroll
  for (int r = 0; r < 8; ++r) hs[w][hi8 + r][m] = tanhf(acc[r] + bias1);
  __syncthreads();

  v2f a2[4];
#pragma unroll
  for (int kb = 0; kb < Hh; kb += 4) {
    a2[kb >> 2].x = hs[w][m][kb + off];
    a2[kb >> 2].y = hs[w][m][kb + off + 1];
  }

  // ---- stage 2: 16x16 @ 16x64 (+ bias) into LDS tile (stride 68) ----
  __shared__ float ob[8][16 * 68];
  for (int nt = 0; nt < 4; ++nt) {
    const int nc = nt * 16 + m;
    const float bias2 = be2[nc];
    v8f c;
#pragma unroll
    for (int r = 0; r < 8; ++r) c[r] = bias2;
#pragma unroll
    for (int kb = 0; kb < Hh; kb += 4) {
      v2f b; b.x = we2[(kb + off) * Ee + nc]; b.y = we2[(kb + off + 1) * Ee + nc];
      c = wmma_f32(a2[kb >> 2], b, c);
    }
#pragma unroll
    for (int r = 0; r < 8; ++r) ob[w][(hi8 + r) * 68 + nc] = c[r];
  }
  __syncthreads();

  // coalesced write-out: 16 rows x 64 floats = 4 KB contiguous per wave
  float* dst = de + ((size_t)s * NE + (size_t)et * 16) * Ee;
#pragma unroll
  for (int i = 0; i < 8; ++i) {
    const int flat = i * 128 + lane * 4;         // 512 B per instruction
    const int row  = flat >> 6;
    const int col  = flat & 63;
    float4 v = *(const float4*)&ob[w][row * 68 + col];
    *(float4*)(dst + flat) = v;
  }
}

// ---------------------------------------------------------------------------
// Launcher
// Inputs: 0:t 1:v0 2:e0 3:wv1 4:bv1 5:wv2 6:bv2 7:we1 8:be1 9:we2 10:be2
//         11:recv_idx 12:send_idx
// Output: [dv (S*N*V) | de (S*NE*E)] fp32
// ---------------------------------------------------------------------------
extern "C" void kernel_launch(void* const* d_in, const int* in_sizes, int n_in,
                              void* d_out, int out_size, void* d_ws, size_t ws_size,
                              hipStream_t stream) {
  (void)in_sizes; (void)n_in; (void)out_size; (void)d_ws; (void)ws_size;
  const float* v0  = (const float*)d_in[1];
  const float* e0  = (const float*)d_in[2];
  const float* wv1 = (const float*)d_in[3];
  const float* bv1 = (const float*)d_in[4];
  const float* wv2 = (const float*)d_in[5];
  const float* bv2 = (const float*)d_in[6];
  const float* we1 = (const float*)d_in[7];
  const float* be1 = (const float*)d_in[8];
  const float* we2 = (const float*)d_in[9];
  const float* be2 = (const float*)d_in[10];
  const int* recv  = (const int*)d_in[11];
  const int* send  = (const int*)d_in[12];

  float* dv = (float*)d_out;                       // also holds ev between k1/k2
  float* de = dv + (size_t)Ss * Nn * Vv;

  k_sum_e0<<<Ss * Nn, 256, 0, stream>>>(e0, dv);
  k_node_mlp<<<TILES_NODE / 8, 256, 0, stream>>>(dv, wv1, bv1, wv2, bv2);
  k_edge_mlp<<<TILES_EDGE / 8, 256, 0, stream>>>(v0, we1, be1, we2, be2,
                                                 recv, send, de);
}